// RawToRgb_49082886259189
// MI455X (gfx1250) — compile-verified
//
#include <hip/hip_runtime.h>
#include <cstdint>

#define H 2048
#define W 2048
#define QH  8                  // quad rows per tile  -> 16 output rows
#define QWT 64                 // quad cols per tile  -> 128 output cols
#define TROWS (2 * QH + 2)     // 18 input rows (1-halo top/bottom)
#define TCOLS (2 * QWT + 2)    // 130 input cols
#define TSTRIDE 132            // padded LDS row stride

typedef __attribute__((ext_vector_type(4))) float f32x4;

// CDNA5 async copy: each lane copies a dword from global vaddr to LDS vdst (GV mode, saddr=off).
// Tracked by ASYNCcnt; drained with s_wait_asynccnt before the barrier.
__device__ __forceinline__ void async_ld_f32(uint32_t lds_byte_off, const float* gaddr) {
    asm volatile("global_load_async_to_lds_b32 %0, %1, off"
                 :: "v"(lds_byte_off),
                    "v"((unsigned long long)(uintptr_t)gaddr)
                 : "memory");
}

__global__ __launch_bounds__(256) void RawToRgb_demosaic(const float* __restrict__ img,
                                                         float* __restrict__ out) {
    __shared__ float tile[TROWS][TSTRIDE];

    const int t     = threadIdx.x;     // 0..255 (8 wave32s)
    const int tileX = blockIdx.x;      // 0..15
    const int tileY = blockIdx.y;      // 0..127
    const int b     = blockIdx.z;

    const int M0 = tileY * QH;
    const int N0 = tileX * QWT;
    const float* src = img + (size_t)b * (size_t)(H * W);

    // ---- Stage rows [2*M0-1 .. 2*M0+16] x cols [2*N0-1 .. 2*N0+128] into LDS.
    // Boundary remap (-1 -> 1, 2048 -> 2046) serves BOTH the clamp-pad (R/B)
    // and reflect-pad (G) semantics — they coincide, proven in round 0.
    const int gy0 = 2 * M0 - 1;
    const int gx0 = 2 * N0 - 1;
    for (int i = t; i < TROWS * TCOLS; i += 256) {
        const int tr = i / TCOLS;
        const int tc = i - tr * TCOLS;
        int gy = gy0 + tr;  gy = (gy < 0) ? 1 : ((gy >= H) ? (H - 2) : gy);
        int gx = gx0 + tc;  gx = (gx < 0) ? 1 : ((gx >= W) ? (W - 2) : gx);
        async_ld_f32((uint32_t)(uintptr_t)(&tile[tr][tc]),
                     src + (size_t)gy * W + gx);
    }
    asm volatile("s_wait_asynccnt 0x0" ::: "memory");
    __syncthreads();

    // ---- Two horizontal 2x2 quads per thread, from a 4x6 LDS window.
    const int qp  = t & 31;            // quad-pair index 0..31
    const int qy  = t >> 5;            // 0..7
    const int tr0 = 2 * qy;
    const int tc0 = 4 * qp;

    float v[4][6];
#pragma unroll
    for (int r = 0; r < 4; ++r)
#pragma unroll
        for (int c = 0; c < 6; ++c)
            v[r][c] = tile[tr0 + r][tc0 + c];

    // Rows: v[0]=img[2m-1], v[1]=img[2m], v[2]=img[2m+1], v[3]=img[2m+2]
    // Cols: c=0 -> x=2n0-1 ... c=5 -> x=2n0+4   (quad A cols 0..3, quad B cols 2..5)
    // Red (sites at odd/odd)
    f32x4 r0 = { 0.25f * (v[0][0] + v[0][2] + v[2][0] + v[2][2]),
                 0.5f  * (v[0][2] + v[2][2]),
                 0.25f * (v[0][2] + v[0][4] + v[2][2] + v[2][4]),
                 0.5f  * (v[0][4] + v[2][4]) };
    f32x4 r1 = { 0.5f  * (v[2][0] + v[2][2]),  v[2][2],
                 0.5f  * (v[2][2] + v[2][4]),  v[2][4] };
    // Green (exact at green sites, 4-neighbor reflect-avg elsewhere)
    f32x4 g0 = { 0.25f * (v[0][1] + v[2][1] + v[1][0] + v[1][2]),  v[1][2],
                 0.25f * (v[0][3] + v[2][3] + v[1][2] + v[1][4]),  v[1][4] };
    f32x4 g1 = { v[2][1],  0.25f * (v[1][2] + v[3][2] + v[2][1] + v[2][3]),
                 v[2][3],  0.25f * (v[1][4] + v[3][4] + v[2][3] + v[2][5]) };
    // Blue (sites at even/even)
    f32x4 b0 = { v[1][1],  0.5f * (v[1][1] + v[1][3]),
                 v[1][3],  0.5f * (v[1][3] + v[1][5]) };
    f32x4 b1 = { 0.5f  * (v[1][1] + v[3][1]),
                 0.25f * (v[1][1] + v[1][3] + v[3][1] + v[3][3]),
                 0.5f  * (v[1][3] + v[3][3]),
                 0.25f * (v[1][3] + v[1][5] + v[3][3] + v[3][5]) };

    const int y0 = 2 * (M0 + qy);
    const int x0 = 2 * N0 + 4 * qp;            // multiple of 4 -> 16B aligned
    const size_t plane = (size_t)H * (size_t)W;
    float* pr = out + (size_t)(b * 3) * plane + (size_t)y0 * W + x0;
    float* pg = pr + plane;
    float* pb = pg + plane;

    // Streaming 384 MB output -> NT stores keep the 128 MB input resident in
    // the 192 MB L2 for cross-tile halo reuse.
    __builtin_nontemporal_store(r0, (f32x4*)pr);
    __builtin_nontemporal_store(r1, (f32x4*)(pr + W));
    __builtin_nontemporal_store(g0, (f32x4*)pg);
    __builtin_nontemporal_store(g1, (f32x4*)(pg + W));
    __builtin_nontemporal_store(b0, (f32x4*)pb);
    __builtin_nontemporal_store(b1, (f32x4*)(pb + W));
}

extern "C" void kernel_launch(void* const* d_in, const int* in_sizes, int n_in,
                              void* d_out, int out_size, void* d_ws, size_t ws_size,
                              hipStream_t stream) {
    const float* img = (const float*)d_in[0];
    float* out = (float*)d_out;
    const int B = in_sizes[0] / (H * W);                 // 8
    dim3 grid(W / (2 * QWT), H / (2 * QH), B);           // (16, 128, 8)
    RawToRgb_demosaic<<<grid, 256, 0, stream>>>(img, out);
}